// GAT_22101901705840
// MI455X (gfx1250) — compile-verified
//
#include <hip/hip_runtime.h>
#include <hip/hip_bf16.h>

typedef float v2f __attribute__((ext_vector_type(2)));
typedef float v8f __attribute__((ext_vector_type(8)));

// ---- monotone encoding for float atomic-max via unsigned atomicMax ----
__device__ __forceinline__ unsigned enc_f32(float f) {
    unsigned u = __float_as_uint(f);
    return (u & 0x80000000u) ? ~u : (u | 0x80000000u);
}
__device__ __forceinline__ float dec_f32(unsigned u) {
    return __uint_as_float((u & 0x80000000u) ? (u & 0x7FFFFFFFu) : ~u);
}

// ---- kernel 1: fold heads of attention params: w_src[64], w_dst[64], bias ----
__global__ void k_params(const float* __restrict__ attn_w,
                         const float* __restrict__ attn_b,
                         float* __restrict__ hdr) {
    int t = threadIdx.x;
    if (t < 64) {
        float a = 0.f, b = 0.f;
        #pragma unroll
        for (int hh = 0; hh < 4; ++hh) {
            a += attn_w[hh * 128 + t];
            b += attn_w[hh * 128 + 64 + t];
        }
        hdr[t] = a;
        hdr[64 + t] = b;
    } else if (t == 64) {
        float s = 0.f;
        #pragma unroll
        for (int hh = 0; hh < 4; ++hh) s += attn_b[hh];
        hdr[128] = s;
    }
}

// ---- kernel 2: init scratch (agg=0 via b128 stores, maxenc=enc(-inf), denom=0) ----
__global__ void k_init(unsigned* __restrict__ maxenc, float* __restrict__ denom,
                       float4* __restrict__ agg4, int N) {
    int i = blockIdx.x * blockDim.x + threadIdx.x;
    if (i < N * 16) agg4[i] = make_float4(0.f, 0.f, 0.f, 0.f);  // N*64 floats
    if (i < N) { maxenc[i] = 0x007FFFFFu; denom[i] = 0.f; }
}

// ---- kernel 3: per-node scalar scores (wave per node, lane owns float2) ----
__global__ void k_scores(const float* __restrict__ h, const float* __restrict__ hdr,
                         float* __restrict__ s_src, float* __restrict__ s_dst, int N) {
    int wid  = (blockIdx.x * blockDim.x + threadIdx.x) >> 5;
    int lane = threadIdx.x & 31;
    if (wid >= N) return;
    float2 hv = *(const float2*)(h + (size_t)wid * 64 + lane * 2);
    float2 wa = *(const float2*)(hdr + lane * 2);        // w_src
    float2 wb = *(const float2*)(hdr + 64 + lane * 2);   // w_dst
    float ps = hv.x * wa.x + hv.y * wa.y;
    float pd = hv.x * wb.x + hv.y * wb.y;
    #pragma unroll
    for (int m = 16; m; m >>= 1) {
        ps += __shfl_xor(ps, m, 32);
        pd += __shfl_xor(pd, m, 32);
    }
    if (lane == 0) { s_src[wid] = ps; s_dst[wid] = pd; }
}

// ---- kernel 4: segment max of edge logits over dst ----
__global__ void k_edge_max(const int* __restrict__ ei,
                           const float* __restrict__ s_src,
                           const float* __restrict__ s_dst,
                           const float* __restrict__ hdr,
                           unsigned* __restrict__ maxenc, int E) {
    int e = blockIdx.x * blockDim.x + threadIdx.x;
    if (e >= E) return;
    int s = ei[e], d = ei[E + e];
    float v = s_src[s] + s_dst[d] + hdr[128];
    atomicMax(&maxenc[d], enc_f32(v));
}

// ---- kernel 5: ex = exp(e - m[dst]); denom[dst] += ex ----
__global__ void k_edge_exp(const int* __restrict__ ei,
                           const float* __restrict__ s_src,
                           const float* __restrict__ s_dst,
                           const float* __restrict__ hdr,
                           const unsigned* __restrict__ maxenc,
                           float* __restrict__ exb,
                           float* __restrict__ denom, int E) {
    int e = blockIdx.x * blockDim.x + threadIdx.x;
    if (e >= E) return;
    int s = ei[e], d = ei[E + e];
    float v = s_src[s] + s_dst[d] + hdr[128];
    float ex = __expf(v - dec_f32(maxenc[d]));
    exb[e] = ex;
    atomicAdd(&denom[d], ex);
}

// ---- kernel 6: weighted scatter agg[dst] += ex * h[src] (wave per edge) ----
__global__ void k_scatter(const int* __restrict__ ei,
                          const float* __restrict__ h,
                          const float* __restrict__ exb,
                          float* __restrict__ agg, int E) {
    int w    = (blockIdx.x * blockDim.x + threadIdx.x) >> 5;
    int lane = threadIdx.x & 31;
    if (w >= E) return;
    int s = ei[w], d = ei[E + w];
    float ex = exb[w];
    float2 hv = *(const float2*)(h + (size_t)s * 64 + lane * 2);
    float* ap = agg + (size_t)d * 64 + lane * 2;
    atomicAdd(ap,     ex * hv.x);
    atomicAdd(ap + 1, ex * hv.y);
}

// ---- kernel 7: h_new = denom>0 ? agg/denom : h ; out = h_new @ Wt + b (WMMA f32) ----
// Block: 512 threads = 16 waves. Wave w owns output columns [16w, 16w+16).
// Per node-group of 16: stage h_new[16][64] in LDS, chain 16x V_WMMA_F32_16X16X4_F32.
// Bias is preloaded into the WMMA C accumulator (bias depends only on column).
__global__ void __launch_bounds__(512)
k_proj_wmma(const float* __restrict__ h,
            const float* __restrict__ agg,
            const float* __restrict__ denom,
            const float* __restrict__ fc_w,
            const float* __restrict__ fc_b,
            float* __restrict__ out, int N) {
    __shared__ float hn[16 * 66];           // padded stride: bank-friendly, 8B-aligned float2
    const int wave = threadIdx.x >> 5;      // 0..15
    const int lane = threadIdx.x & 31;
    const int hi   = lane >> 4;             // 0/1 half-wave
    const int lo   = lane & 15;
    const int col  = wave * 16 + lo;        // 0..255  (head = col>>6, o = col&63)

    // Preload this wave's B tiles once: Wt[k][col] = fc_w[head][o][k]
    const float* wp = fc_w + (size_t)col * 64;  // (head*64+o)*64 == col*64
    float2 breg[16];
    #pragma unroll
    for (int kk = 0; kk < 16; ++kk) {
        int k = kk * 4 + 2 * hi;
        breg[kk] = make_float2(wp[k], wp[k + 1]);
    }
    const float fb = fc_b[col];

    const int ngroups = (N + 15) >> 4;
    for (int g = blockIdx.x; g < ngroups; g += gridDim.x) {
        __syncthreads();  // all waves done reading previous tile
        // stage h_new rows for 16 nodes
        for (int i = threadIdx.x; i < 16 * 64; i += 512) {
            int m = i >> 6, dcol = i & 63;
            int node = g * 16 + m;
            float v = 0.f;
            if (node < N) {
                float dn = denom[node];
                v = (dn > 0.f) ? agg[(size_t)node * 64 + dcol] / dn
                               : h[(size_t)node * 64 + dcol];
            }
            hn[m * 66 + dcol] = v;
        }
        __syncthreads();

        // C starts at the per-column bias: D = A*B + bias
        v8f c = { fb, fb, fb, fb, fb, fb, fb, fb };
        #pragma unroll
        for (int kk = 0; kk < 16; ++kk) {
            int k = kk * 4 + 2 * hi;
            v2f a; a.x = hn[lo * 66 + k]; a.y = hn[lo * 66 + k + 1];
            v2f b; b.x = breg[kk].x;      b.y = breg[kk].y;
            c = __builtin_amdgcn_wmma_f32_16x16x4_f32(
                    false, a, false, b, (short)0, c, false, false);
        }

        float* op = out + ((size_t)g * 16 + hi * 8) * 256 + col;
        if (g * 16 + 16 <= N) {
            // full group: uniform, unguarded stores
            #pragma unroll
            for (int i = 0; i < 8; ++i) op[(size_t)i * 256] = c[i];
        } else {
            #pragma unroll
            for (int i = 0; i < 8; ++i) {
                int node = g * 16 + hi * 8 + i;
                if (node < N) op[(size_t)i * 256] = c[i];
            }
        }
    }
}

extern "C" void kernel_launch(void* const* d_in, const int* in_sizes, int n_in,
                              void* d_out, int out_size, void* d_ws, size_t ws_size,
                              hipStream_t stream) {
    const float* h      = (const float*)d_in[0];
    const float* attn_w = (const float*)d_in[1];
    const float* attn_b = (const float*)d_in[2];
    const float* fc_w   = (const float*)d_in[3];
    const float* fc_b   = (const float*)d_in[4];
    const int*   ei     = (const int*)d_in[5];

    const int N = in_sizes[0] / 64;
    const int E = in_sizes[5] / 2;

    // workspace layout (floats)
    float*    wsf    = (float*)d_ws;
    float*    hdr    = wsf;                         // 256 floats (w_src|w_dst|bias)
    float*    s_src  = wsf + 256;                   // N
    float*    s_dst  = s_src + N;                   // N
    unsigned* maxenc = (unsigned*)(s_dst + N);      // N
    float*    denom  = wsf + 256 + 3 * (size_t)N;   // N
    float*    exb    = denom + N;                   // E
    float*    agg    = exb + E;                     // N*64  (16-byte aligned: offsets all mult of 4 floats)

    float* out = (float*)d_out;

    k_params<<<1, 128, 0, stream>>>(attn_w, attn_b, hdr);
    k_init<<<(N * 16 + 255) / 256, 256, 0, stream>>>(maxenc, denom, (float4*)agg, N);
    k_scores<<<(N + 7) / 8, 256, 0, stream>>>(h, hdr, s_src, s_dst, N);
    k_edge_max<<<(E + 255) / 256, 256, 0, stream>>>(ei, s_src, s_dst, hdr, maxenc, E);
    k_edge_exp<<<(E + 255) / 256, 256, 0, stream>>>(ei, s_src, s_dst, hdr, maxenc, exb, denom, E);
    k_scatter<<<(E + 7) / 8, 256, 0, stream>>>(ei, h, exb, agg, E);
    k_proj_wmma<<<2048, 512, 0, stream>>>(h, agg, denom, fc_w, fc_b, out, N);
}